// Constrained_LCBlock_60988535603753
// MI455X (gfx1250) — compile-verified
//
#include <hip/hip_runtime.h>
#include <hip/hip_bf16.h>

typedef float v2f __attribute__((ext_vector_type(2)));
typedef float v8f __attribute__((ext_vector_type(8)));
typedef unsigned int v4u __attribute__((ext_vector_type(4)));
typedef int v8i __attribute__((ext_vector_type(8)));
typedef int v4i __attribute__((ext_vector_type(4)));

#define HH 64
#define WW 64
#define CI 64
#define CO 64
#define KPRIME 576      // CI * 3 * 3
#define CIB 4           // ci per K-chunk
#define KC  (CIB * 9)   // 36 k-values per chunk
#define NCH (CI / CIB)  // 16 chunks
#define RP  36          // TDM-packed LDS row stride (floats): conflict-free (see analysis)
#define NSTEP (KC / 4)  // 9 wmma k-steps per chunk

__global__ __launch_bounds__(256, 1)
void lc_wmma_tdm_kernel(const float* __restrict__ x,
                        const float* __restrict__ wts,
                        float* __restrict__ out)
{
    // Double buffers: weights tile (64 co x 36 k, TDM-written) and patch tile (64 b x 36 k)
    __shared__ float Wl[2][CO * RP];
    __shared__ float Pl[2][64 * RP];

    const int w = blockIdx.x;
    const int h = blockIdx.y;
    const int t = threadIdx.x;
    const int lane = t & 31;
    const int wv   = t >> 5;

    // ---- patch staging mapping: b = t>>2, local ci = t&3, 3x3 window each
    const int row = t >> 2;
    const int q   = t & 3;
    const int y0 = h - 1, xw0 = w - 1;

    // ---- compute mapping: wave -> (m-tile, two n-tiles)
    const int m0 = (wv & 3) * 16;
    const int n0 = (wv >> 2) * 32;
    const int lr = lane & 15;
    const int lh = lane >> 4;
    const int aoff  = (m0 + lr) * RP + 2 * lh;
    const int boff0 = (n0 + lr) * RP + 2 * lh;
    const int boff1 = boff0 + 16 * RP;

    v8f acc0a = {}, acc0b = {}, acc1a = {}, acc1b = {};
    float preg[9];

    // ---- TDM setup (all uniform values -> SGPRs)
    // Weight tile for this (h,w): 64 rows (co) x 36 fp32, global row stride 576 fp32.
    const unsigned long long wloc =
        (unsigned long long)(const void*)wts +
        (unsigned long long)((h * WW + w) * CO) * (KPRIME * 4ull);
    // LDS byte offset of Wl: LDS aperture occupies only the high 32 bits of a flat
    // address (ISA aperture check), so truncation yields the raw LDS offset.
    const unsigned ldsW0 = (unsigned)__builtin_amdgcn_readfirstlane(
        (int)(unsigned long long)(const void*)&Wl[0][0]);

    auto tdm_issue = [&](int c) {
        const unsigned long long ga = wloc + (unsigned long long)c * (KC * 4ull);
        const unsigned lds_addr = ldsW0 + (unsigned)(c & 1) * (CO * RP * 4u);
        v4u g0;
        g0[0] = 1u;                                   // count=1, no gather
        g0[1] = lds_addr;                             // lds_addr [63:32]
        g0[2] = (unsigned)ga;                         // global_addr [95:64]
        g0[3] = ((unsigned)(ga >> 32) & 0x01FFFFFFu)  // global_addr [120:96]
                | (2u << 30);                         // type = 2 ("image")
        v8i g1;
        g1[0] = (int)(2u << 16);     // workgroup_mask=0, data_size=2 (4 bytes)
        g1[1] = (int)(36u << 16);    // atomic_barrier=0, tensor_dim0[15:0]=36
        g1[2] = (int)(64u << 16);    // tensor_dim0[31:16]=0, tensor_dim1[15:0]=64
        g1[3] = (int)(36u << 16);    // tensor_dim1[31:16]=0, tile_dim0=36
        g1[4] = 64;                  // tile_dim1=64, tile_dim2=0
        g1[5] = 576;                 // tensor_dim0_stride[31:0]=576
        g1[6] = 0;                   // stride hi, tensor_dim1_stride lo
        g1[7] = 0;
        const v4i gz4 = {0, 0, 0, 0};          // groups 2/3 unused (2D tensor)
        const v8i gz8 = {0, 0, 0, 0, 0, 0, 0, 0};
        // amdgpu-toolchain (clang-23) 6-arg form: (g0, g1, g2, g3, v8i, cpol)
        __builtin_amdgcn_tensor_load_to_lds(g0, g1, gz4, gz4, gz8, 0);
    };

    auto stage_load = [&](int c) {
        // patch: b = row, ci = c*CIB + q, 3x3 window with zero pad (x is L2-resident)
        const int ciG = c * CIB + q;
        const float* xp = x + ((long)(row * CI + ciG)) * (HH * WW);
        #pragma unroll
        for (int dy = 0; dy < 3; ++dy) {
            const int yy = y0 + dy;
            #pragma unroll
            for (int dx = 0; dx < 3; ++dx) {
                const int xx = xw0 + dx;
                const bool ok = (yy >= 0) & (yy < HH) & (xx >= 0) & (xx < WW);
                preg[dy * 3 + dx] = ok ? xp[yy * WW + xx] : 0.0f;
            }
        }
    };

    auto stage_store = [&](int buf) {
        float* pl = &Pl[buf][row * RP + q * 9];
        #pragma unroll
        for (int i = 0; i < 9; ++i) pl[i] = preg[i];
    };

    auto compute = [&](int buf) {
        const float* Wb = &Wl[buf][0];
        const float* Pb = &Pl[buf][0];
        #pragma unroll
        for (int ks = 0; ks < NSTEP; ++ks) {
            const int k0 = ks * 4;
            v2f a  = *(const v2f*)(Wb + aoff  + k0);
            v2f b0 = *(const v2f*)(Pb + boff0 + k0);
            v2f b1 = *(const v2f*)(Pb + boff1 + k0);
            if (ks & 1) {
                acc0b = __builtin_amdgcn_wmma_f32_16x16x4_f32(false, a, false, b0,
                                                              (short)0, acc0b, false, false);
                acc1b = __builtin_amdgcn_wmma_f32_16x16x4_f32(false, a, false, b1,
                                                              (short)0, acc1b, false, false);
            } else {
                acc0a = __builtin_amdgcn_wmma_f32_16x16x4_f32(false, a, false, b0,
                                                              (short)0, acc0a, false, false);
                acc1a = __builtin_amdgcn_wmma_f32_16x16x4_f32(false, a, false, b1,
                                                              (short)0, acc1a, false, false);
            }
        }
    };

    // ---- pipelined main loop: TDM DMA for chunk c+1 overlaps WMMA on chunk c
    if (t < 32) tdm_issue(0);                     // wave 0 issues DMA (TDM ignores EXEC)
    stage_load(0);
    stage_store(0);
    if (t < 32) __builtin_amdgcn_s_wait_tensorcnt(0);
    __syncthreads();

    for (int c = 0; c < NCH; ++c) {
        if (c + 1 < NCH) {
            if (t < 32) tdm_issue(c + 1);         // DMA into the other buffer
            stage_load(c + 1);                    // patch gather into registers
        }
        compute(c & 1);                           // 18 WMMAs from LDS
        if (c + 1 < NCH) {
            stage_store((c + 1) & 1);             // patch registers -> LDS
            if (t < 32) __builtin_amdgcn_s_wait_tensorcnt(0);
        }
        __syncthreads();
    }

    // ---- epilogue: combine chains, ReLU, store (out surface is L2-resident)
    v8f r0 = acc0a + acc0b;
    v8f r1 = acc1a + acc1b;
    const int bN0 = n0 + lr;
    const int bN1 = bN0 + 16;
    #pragma unroll
    for (int v = 0; v < 8; ++v) {
        const int co = m0 + 8 * lh + v;           // D layout: VGPR v + lane-half -> M
        float v0 = r0[v]; v0 = v0 > 0.0f ? v0 : 0.0f;
        float v1 = r1[v]; v1 = v1 > 0.0f ? v1 : 0.0f;
        out[((bN0 * CO + co) * HH + h) * WW + w] = v0;
        out[((bN1 * CO + co) * HH + h) * WW + w] = v1;
    }
}

extern "C" void kernel_launch(void* const* d_in, const int* in_sizes, int n_in,
                              void* d_out, int out_size, void* d_ws, size_t ws_size,
                              hipStream_t stream) {
    const float* x   = (const float*)d_in[0];
    const float* wts = (const float*)d_in[1];
    float* out = (float*)d_out;
    dim3 grid(WW, HH);   // one workgroup per spatial site
    lc_wmma_tdm_kernel<<<grid, 256, 0, stream>>>(x, wts, out);
}